// GNNCaptioner_4157528342613
// MI455X (gfx1250) — compile-verified
//
#include <hip/hip_runtime.h>
#include <math.h>
#include <stdint.h>

// ---------------------------------------------------------------------------
// Types for CDNA5 WMMA f32 16x16x4
// ---------------------------------------------------------------------------
typedef float v2f __attribute__((ext_vector_type(2)));
typedef float v8f __attribute__((ext_vector_type(8)));

#define LRELU_SLOPE 0.2f

// ---------------------------------------------------------------------------
// Async copy helpers (CDNA5 GLOBAL_LOAD_ASYNC_TO_LDS, tracked by ASYNCcnt).
// LDS byte offset = low 32 bits of the generic pointer (aperture in high bits).
// ---------------------------------------------------------------------------
__device__ __forceinline__ void async_copy_b32(const float* gp, const float* ldsp) {
    unsigned ldsoff = (unsigned)(uintptr_t)ldsp;
    unsigned long long ga = (unsigned long long)(uintptr_t)gp;
    asm volatile("global_load_async_to_lds_b32 %0, %1, off"
                 :: "v"(ldsoff), "v"(ga) : "memory");
}
__device__ __forceinline__ void async_wait_all() {
    asm volatile("s_wait_asynccnt 0" ::: "memory");
}

// ---------------------------------------------------------------------------
// Order-preserving float <-> uint key (for atomicMax-based segment max)
// ---------------------------------------------------------------------------
__device__ __forceinline__ unsigned fkey(float f) {
    unsigned u = __float_as_uint(f);
    return (u & 0x80000000u) ? ~u : (u | 0x80000000u);
}
__device__ __forceinline__ float funkey(unsigned k) {
    unsigned u = (k & 0x80000000u) ? (k & 0x7FFFFFFFu) : ~k;
    return __uint_as_float(u);
}

// ---------------------------------------------------------------------------
// Generic fp32 WMMA GEMM: C[M,N] = A[M,K] * B + bias(col)
//   TRANSB=0 : B is [K,N] row-major
//   TRANSB=1 : B is [N,K] row-major (i.e. use B^T)
// Block = 256 threads (8 waves), block tile 64(M) x 32(N),
// each wave computes one 16x16 f32 tile via v_wmma_f32_16x16x4_f32.
// Interior tiles stage A/B via global_load_async_to_lds_b32 (ASYNCcnt path);
// ragged boundary tiles use the predicated zero-fill path.
// ---------------------------------------------------------------------------
template <bool TRANSB>
__global__ __launch_bounds__(256) void gemm_wmma_f32(
    const float* __restrict__ A, const float* __restrict__ B,
    float* __restrict__ C, const float* __restrict__ bias,
    int M, int N, int K)
{
    __shared__ float As[16][64];   // [k][m]
    __shared__ float Bs[16][32];   // [k][n]

    const int tid   = threadIdx.x;
    const int lane  = tid & 31;
    const int wave  = tid >> 5;
    const int wm    = wave & 3;        // wave M tile 0..3
    const int wn    = wave >> 2;       // wave N tile 0..1
    const int blockM = blockIdx.y * 64;
    const int blockN = blockIdx.x * 32;

    const int mloc  = (lane & 15) + wm * 16;      // A fragment: M index
    const int nloc  = (lane & 15) + wn * 16;      // B fragment: N index
    const int khalf = (lane >> 4) << 1;           // 0 or 2: K sub-offset

    const bool interiorA = (blockM + 64 <= M);    // block-uniform
    const bool interiorB = (blockN + 32 <= N);    // block-uniform

    v8f acc = {0.f, 0.f, 0.f, 0.f, 0.f, 0.f, 0.f, 0.f};

    for (int kt = 0; kt < K; kt += 16) {
        // ---- stage A tile: 64x16 (4 elems/thread, uniform trip count) ----
        if (interiorA) {
            #pragma unroll
            for (int j = 0; j < 4; ++j) {
                int i = tid + j * 256;
                int m = i >> 4, k = i & 15;
                async_copy_b32(&A[(size_t)(blockM + m) * K + kt + k], &As[k][m]);
            }
        } else {
            for (int i = tid; i < 64 * 16; i += 256) {
                int m = i >> 4, k = i & 15;
                int row = blockM + m;
                As[k][m] = (row < M) ? A[(size_t)row * K + kt + k] : 0.f;
            }
        }
        // ---- stage B tile: 16x32 (2 elems/thread) ----
        if (interiorB) {
            #pragma unroll
            for (int j = 0; j < 2; ++j) {
                int i = tid + j * 256;
                int n = i & 31, k = i >> 5;
                const float* gp = TRANSB ? &B[(size_t)(blockN + n) * K + kt + k]
                                         : &B[(size_t)(kt + k) * N + blockN + n];
                async_copy_b32(gp, &Bs[k][n]);
            }
        } else {
            for (int i = tid; i < 16 * 32; i += 256) {
                int n = i & 31, k = i >> 5;
                int col = blockN + n;
                float v = 0.f;
                if (col < N)
                    v = TRANSB ? B[(size_t)col * K + kt + k]
                               : B[(size_t)(kt + k) * N + col];
                Bs[k][n] = v;
            }
        }
        // speculative prefetch of the next A tile -> global_prefetch_b8
        if (kt + 16 < K) {
            int prow = blockM + (tid >> 2);
            if (prow < M)
                __builtin_prefetch(&A[(size_t)prow * K + kt + 16], 0, 1);
        }
        async_wait_all();          // ASYNCcnt == 0: LDS writes landed
        __syncthreads();

        // ---- 4 x wmma_f32_16x16x4_f32 ----
        #pragma unroll
        for (int kk = 0; kk < 4; ++kk) {
            int k0 = kk * 4 + khalf;
            v2f a, b;
            a[0] = As[k0][mloc];     a[1] = As[k0 + 1][mloc];
            b[0] = Bs[k0][nloc];     b[1] = Bs[k0 + 1][nloc];
            acc = __builtin_amdgcn_wmma_f32_16x16x4_f32(
                false, a, false, b, (short)0, acc, false, false);
        }
        __syncthreads();
    }

    // ---- store C fragment: vgpr r -> M = r + 8*(lane>=16), N = lane&15 ----
    #pragma unroll
    for (int r = 0; r < 8; ++r) {
        int row = blockM + wm * 16 + r + ((lane >> 4) << 3);
        int col = blockN + wn * 16 + (lane & 15);
        if (row < M && col < N) {
            float v = acc[r];
            if (bias) v += bias[col];
            C[(size_t)row * N + col] = v;
        }
    }
}

// ---------------------------------------------------------------------------
// Per-node attention dot products: es[n] = h[n].a_src, ed[n] = h[n].a_dst
// (one wave per node)
// ---------------------------------------------------------------------------
__global__ void node_dots(const float* __restrict__ h,
                          const float* __restrict__ a_s,
                          const float* __restrict__ a_d,
                          float* __restrict__ es, float* __restrict__ ed,
                          int N, int HID)
{
    const int lane = threadIdx.x & 31;
    const int node = blockIdx.x * (blockDim.x >> 5) + (threadIdx.x >> 5);
    if (node >= N) return;
    const float* hp = h + (size_t)node * HID;
    float s1 = 0.f, s2 = 0.f;
    for (int c = lane; c < HID; c += 32) {
        float hv = hp[c];
        s1 += hv * a_s[c];
        s2 += hv * a_d[c];
    }
    #pragma unroll
    for (int off = 16; off > 0; off >>= 1) {
        s1 += __shfl_down(s1, off, 32);
        s2 += __shfl_down(s2, off, 32);
    }
    if (lane == 0) { es[node] = s1; ed[node] = s2; }
}

// ---------------------------------------------------------------------------
// Zero attention accumulators (called every launch: ws is poisoned once)
// ---------------------------------------------------------------------------
__global__ void init_attn(unsigned* __restrict__ mkey, float* __restrict__ denom,
                          float* __restrict__ agg, int N, int HID)
{
    int i = blockIdx.x * blockDim.x + threadIdx.x;
    if (i < N) { mkey[i] = 0u; denom[i] = 0.f; }
    if (i < N * HID) agg[i] = 0.f;
}

// ---------------------------------------------------------------------------
// Edge pass 1: e = leaky_relu(es[src] + ed[dst]); segment max via atomicMax
// ---------------------------------------------------------------------------
__global__ void edge_logits(const int* __restrict__ ei_src, const int* __restrict__ ei_dst,
                            const float* __restrict__ es, const float* __restrict__ ed,
                            float* __restrict__ e, unsigned* __restrict__ mkey,
                            int EN, int E)
{
    int k = blockIdx.x * blockDim.x + threadIdx.x;
    if (k >= EN) return;
    int s = (k < E) ? ei_src[k] : (k - E);
    int d = (k < E) ? ei_dst[k] : (k - E);
    float v = es[s] + ed[d];
    v = (v > 0.f) ? v : LRELU_SLOPE * v;
    e[k] = v;
    atomicMax(&mkey[d], fkey(v));
}

// ---------------------------------------------------------------------------
// Edge pass 2: w = exp(e - m[dst]); denom[dst] += w
// ---------------------------------------------------------------------------
__global__ void edge_weights(const int* __restrict__ ei_dst,
                             const float* __restrict__ e, const unsigned* __restrict__ mkey,
                             float* __restrict__ w, float* __restrict__ denom,
                             int EN, int E)
{
    int k = blockIdx.x * blockDim.x + threadIdx.x;
    if (k >= EN) return;
    int d = (k < E) ? ei_dst[k] : (k - E);
    float m = funkey(mkey[d]);
    float wv = expf(e[k] - m);
    w[k] = wv;
    atomicAdd(&denom[d], wv);
}

// ---------------------------------------------------------------------------
// Edge pass 3: agg[dst] += (w/denom[dst]) * h[src]  (wave per edge)
// ---------------------------------------------------------------------------
__global__ void edge_scatter(const int* __restrict__ ei_src, const int* __restrict__ ei_dst,
                             const float* __restrict__ w, const float* __restrict__ denom,
                             const float* __restrict__ h, float* __restrict__ agg,
                             int EN, int E, int HID)
{
    const int lane = threadIdx.x & 31;
    const int k = blockIdx.x * (blockDim.x >> 5) + (threadIdx.x >> 5);
    if (k >= EN) return;
    int s = (k < E) ? ei_src[k] : (k - E);
    int d = (k < E) ? ei_dst[k] : (k - E);
    float alpha = w[k] / denom[d];
    const float* hp = h + (size_t)s * HID;
    float* ag = agg + (size_t)d * HID;
    for (int c = lane; c < HID; c += 32)
        atomicAdd(&ag[c], alpha * hp[c]);
}

// ---------------------------------------------------------------------------
// Node finalize: out = agg + bias (+ optional relu)
// ---------------------------------------------------------------------------
__global__ void finalize_nodes(const float* __restrict__ agg, const float* __restrict__ bias,
                               float* __restrict__ out, int N, int HID, int relu)
{
    int i = blockIdx.x * blockDim.x + threadIdx.x;
    if (i >= N * HID) return;
    int c = i % HID;
    float v = agg[i] + bias[c];
    if (relu) v = fmaxf(v, 0.f);
    out[i] = v;
}

__global__ void bias_sum(const float* __restrict__ a, const float* __restrict__ b,
                         float* __restrict__ o, int n)
{
    int i = blockIdx.x * blockDim.x + threadIdx.x;
    if (i < n) o[i] = a[i] + b[i];
}

// ---------------------------------------------------------------------------
// LSTM scan: one persistent block of 4H (=1024) threads, T sequential steps.
// h lives in LDS; c lives in registers of threads [0,H).
// gate order i,f,g,o (torch convention).
// ---------------------------------------------------------------------------
__global__ __launch_bounds__(1024) void lstm_scan(
    const float* __restrict__ pre,   // [T, 4H]
    const float* __restrict__ W_hh,  // [4H, H]
    float* __restrict__ hs,          // [T, H]
    int T, int H)
{
    __shared__ __align__(16) float h_s[256];
    __shared__ float g_s[1024];
    const int tid = threadIdx.x;
    const int H4 = 4 * H;

    if (tid < H) h_s[tid] = 0.f;
    float c = 0.f;
    __syncthreads();

    const float* wrow = W_hh + (size_t)tid * H;

    for (int t = 0; t < T; ++t) {
        float acc = pre[(size_t)t * H4 + tid];
        #pragma unroll 8
        for (int k = 0; k < 256; k += 4) {
            float4 wv = *reinterpret_cast<const float4*>(wrow + k);
            float4 hv = *reinterpret_cast<const float4*>(&h_s[k]);
            acc += wv.x * hv.x + wv.y * hv.y + wv.z * hv.z + wv.w * hv.w;
        }
        g_s[tid] = acc;
        __syncthreads();
        if (tid < H) {
            float gi = 1.f / (1.f + expf(-g_s[tid]));
            float gf = 1.f / (1.f + expf(-g_s[H + tid]));
            float gg = tanhf(g_s[2 * H + tid]);
            float go = 1.f / (1.f + expf(-g_s[3 * H + tid]));
            c = gf * c + gi * gg;
            float hn = go * tanhf(c);
            h_s[tid] = hn;
            hs[(size_t)t * H + tid] = hn;
        }
        __syncthreads();
    }
}

// ---------------------------------------------------------------------------
// Host launcher
// ---------------------------------------------------------------------------
extern "C" void kernel_launch(void* const* d_in, const int* in_sizes, int n_in,
                              void* d_out, int out_size, void* d_ws, size_t ws_size,
                              hipStream_t stream)
{
    // inputs per setup_inputs() order
    const float* x      = (const float*)d_in[0];
    const float* W1     = (const float*)d_in[1];
    const float* a_src1 = (const float*)d_in[2];
    const float* a_dst1 = (const float*)d_in[3];
    const float* b1     = (const float*)d_in[4];
    const float* W2     = (const float*)d_in[5];
    const float* a_src2 = (const float*)d_in[6];
    const float* a_dst2 = (const float*)d_in[7];
    const float* b2     = (const float*)d_in[8];
    const float* W_ih   = (const float*)d_in[9];
    const float* W_hh   = (const float*)d_in[10];
    const float* b_ih   = (const float*)d_in[11];
    const float* b_hh   = (const float*)d_in[12];
    const float* Wfc    = (const float*)d_in[13];
    const float* bfc    = (const float*)d_in[14];
    const int*   ei     = (const int*)d_in[15];   // [2, E]

    const int HID    = in_sizes[2];               // 256
    const int IN_DIM = in_sizes[1] / HID;         // 512
    const int N      = in_sizes[0] / IN_DIM;      // 10000
    const int H4     = in_sizes[11];              // 1024
    const int VOCAB  = in_sizes[14];              // 1000
    const int E      = in_sizes[15] / 2;          // 500000
    const int EN     = E + N;                     // edges + self loops
    const int* ei_src = ei;
    const int* ei_dst = ei + E;

    // workspace carve-out
    auto aln = [](size_t v) { return (v + 255) & ~((size_t)255); };
    char* p = (char*)d_ws;
    float* h1    = (float*)p;  p += aln((size_t)N * HID * 4);
    float* h2    = (float*)p;  p += aln((size_t)N * HID * 4);
    float* agg   = (float*)p;  p += aln((size_t)N * HID * 4);
    float* pre   = (float*)p;  p += aln((size_t)N * H4 * 4);
    float* hsbuf = (float*)p;  p += aln((size_t)N * HID * 4);
    float* ebuf  = (float*)p;  p += aln((size_t)EN * 4);
    float* wbuf  = (float*)p;  p += aln((size_t)EN * 4);
    float* es    = (float*)p;  p += aln((size_t)N * 4);
    float* ed    = (float*)p;  p += aln((size_t)N * 4);
    unsigned* mkey = (unsigned*)p; p += aln((size_t)N * 4);
    float* denom = (float*)p;  p += aln((size_t)N * 4);
    float* bsum  = (float*)p;  p += aln((size_t)H4 * 4);

    const dim3 blk(256);
    const dim3 gEdge((EN + 255) / 256);
    const dim3 gEdgeW((EN + 7) / 8);          // 8 waves per block, wave/edge
    const dim3 gNode((N + 7) / 8);            // wave/node
    const dim3 gNH((N * HID + 255) / 256);

    // ===== GAT layer 1 =====
    {
        dim3 g((HID + 31) / 32, (N + 63) / 64);
        gemm_wmma_f32<false><<<g, blk, 0, stream>>>(x, W1, h1, nullptr, N, HID, IN_DIM);
    }
    node_dots<<<gNode, blk, 0, stream>>>(h1, a_src1, a_dst1, es, ed, N, HID);
    init_attn<<<gNH, blk, 0, stream>>>(mkey, denom, agg, N, HID);
    edge_logits<<<gEdge, blk, 0, stream>>>(ei_src, ei_dst, es, ed, ebuf, mkey, EN, E);
    edge_weights<<<gEdge, blk, 0, stream>>>(ei_dst, ebuf, mkey, wbuf, denom, EN, E);
    edge_scatter<<<gEdgeW, blk, 0, stream>>>(ei_src, ei_dst, wbuf, denom, h1, agg, EN, E, HID);
    finalize_nodes<<<gNH, blk, 0, stream>>>(agg, b1, h1, N, HID, /*relu=*/1);

    // ===== GAT layer 2 =====
    {
        dim3 g((HID + 31) / 32, (N + 63) / 64);
        gemm_wmma_f32<false><<<g, blk, 0, stream>>>(h1, W2, h2, nullptr, N, HID, HID);
    }
    node_dots<<<gNode, blk, 0, stream>>>(h2, a_src2, a_dst2, es, ed, N, HID);
    init_attn<<<gNH, blk, 0, stream>>>(mkey, denom, agg, N, HID);
    edge_logits<<<gEdge, blk, 0, stream>>>(ei_src, ei_dst, es, ed, ebuf, mkey, EN, E);
    edge_weights<<<gEdge, blk, 0, stream>>>(ei_dst, ebuf, mkey, wbuf, denom, EN, E);
    edge_scatter<<<gEdgeW, blk, 0, stream>>>(ei_src, ei_dst, wbuf, denom, h2, agg, EN, E, HID);
    finalize_nodes<<<gNH, blk, 0, stream>>>(agg, b2, h2, N, HID, /*relu=*/0);

    // ===== LSTM =====
    bias_sum<<<(H4 + 255) / 256, blk, 0, stream>>>(b_ih, b_hh, bsum, H4);
    {
        dim3 g((H4 + 31) / 32, (N + 63) / 64);
        gemm_wmma_f32<true><<<g, blk, 0, stream>>>(h2, W_ih, pre, bsum, N, H4, HID);
    }
    lstm_scan<<<1, H4, 0, stream>>>(pre, W_hh, hsbuf, N, HID);

    // ===== FC =====
    {
        dim3 g((VOCAB + 31) / 32, (N + 63) / 64);
        gemm_wmma_f32<true><<<g, blk, 0, stream>>>(hsbuf, Wfc, (float*)d_out, bfc, N, VOCAB, HID);
    }
}